// L1_Sobel_Loss_65558380806217
// MI455X (gfx1250) — compile-verified
//
#include <hip/hip_runtime.h>

typedef __attribute__((ext_vector_type(2))) float v2f;
typedef __attribute__((ext_vector_type(8))) float v8f;

#define H 512
#define W 512
#define OH 510
#define OW 510
#define TH 32
#define TW 128
#define HALO_H (TH + 2)   /* 34  */
#define HALO_W (TW + 2)   /* 130 */
#define LDSW 132          /* padded LDS row stride (floats) */
#define NBLOCKS (4 * 16 * 48)

// Wave-level partial reduction via V_WMMA_F32_16X16X4_F32.
// A = all-ones 16x4 (layout-independent: every element is 1.0).
// B = per-lane values (b.x = v, b.y = 0): the 32 lane values occupy 32 of the
// 64 B-matrix slots, the rest are 0. D[m][n] = sum_k B[k][n], so d[0] in lanes
// 0..15 holds the 16 column sums (documented C/D layout: VGPR0 lanes 0-15 ->
// M=0, N=lane). Their sum == sum of all lane values, exactly in f32/RNE.
__device__ __forceinline__ float wave_wmma_colsum(float v) {
  v2f a; a.x = 1.0f; a.y = 1.0f;
  v2f b; b.x = v;    b.y = 0.0f;
  v8f c = {};
  c = __builtin_amdgcn_wmma_f32_16x16x4_f32(false, a, false, b,
                                            (short)0, c, false, false);
  return c[0];
}

// 256-thread (8-wave) block reduction. Returns block total in thread 0.
// EXEC must be all-1s on entry (no divergence at this point).
__device__ __forceinline__ float block_reduce_256(float acc, float* red) {
  const int tid  = threadIdx.x;
  const int wave = tid >> 5;
  const int lane = tid & 31;

  float cs = wave_wmma_colsum(acc);          // lanes 0..15 partition wave total
  if (lane < 16) red[wave * 16 + lane] = cs; // 8 waves * 16 = 128 values
  __syncthreads();

  float total = 0.0f;
  if (wave == 0) {
    float x = red[lane] + red[lane + 32] + red[lane + 64] + red[lane + 96];
    #pragma unroll
    for (int off = 16; off > 0; off >>= 1)
      x += __shfl_down(x, off, 32);
    total = x;
  }
  return total;
}

__global__ void __launch_bounds__(256)
sobel_partial_kernel(const float* __restrict__ src,
                     const float* __restrict__ tgt,
                     float* __restrict__ partial) {
  __shared__ float sA[HALO_H * LDSW];
  __shared__ float sB[HALO_H * LDSW];
  __shared__ float red[128];

  const int tid   = threadIdx.x;
  const int gx0   = blockIdx.x * TW;   // 0..3  -> output x origin
  const int gy0   = blockIdx.y * TH;   // 0..15 -> output y origin
  const int plane = blockIdx.z;        // 0..47 (B*C)
  const size_t base = (size_t)plane * (size_t)(H * W);
  const float* __restrict__ pa = src + base;
  const float* __restrict__ pb = tgt + base;

  // Stage halo tile of both images into LDS (coalesced b32 loads).
  for (int i = tid; i < HALO_H * HALO_W; i += 256) {
    const int r  = i / HALO_W;
    const int cc = i - r * HALO_W;
    const int gy = gy0 + r;
    const int gx = gx0 + cc;
    const bool ok = (gy < H) && (gx < W);
    const size_t off = (size_t)gy * W + (size_t)gx;
    sA[r * LDSW + cc] = ok ? pa[off] : 0.0f;
    sB[r * LDSW + cc] = ok ? pb[off] : 0.0f;
  }
  __syncthreads();

  float acc = 0.0f;
  #pragma unroll 4
  for (int p = tid; p < TH * TW; p += 256) {
    const int oyl = p >> 7;         // p / TW
    const int oxl = p & (TW - 1);   // p % TW
    if ((gy0 + oyl) < OH && (gx0 + oxl) < OW) {
      const float* a0 = &sA[oyl * LDSW + oxl];
      const float* b0 = &sB[oyl * LDSW + oxl];

      float a00 = a0[0],        a01 = a0[1],            a02 = a0[2];
      float a10 = a0[LDSW],                             a12 = a0[LDSW + 2];
      float a20 = a0[2 * LDSW], a21 = a0[2 * LDSW + 1], a22 = a0[2 * LDSW + 2];
      float gxa = (a00 - a02) + 2.0f * (a10 - a12) + (a20 - a22);
      float gya = (a00 + 2.0f * a01 + a02) - (a20 + 2.0f * a21 + a22);
      float ma  = sqrtf(gxa * gxa + gya * gya);

      float c00 = b0[0],        c01 = b0[1],            c02 = b0[2];
      float c10 = b0[LDSW],                             c12 = b0[LDSW + 2];
      float c20 = b0[2 * LDSW], c21 = b0[2 * LDSW + 1], c22 = b0[2 * LDSW + 2];
      float gxb = (c00 - c02) + 2.0f * (c10 - c12) + (c20 - c22);
      float gyb = (c00 + 2.0f * c01 + c02) - (c20 + 2.0f * c21 + c22);
      float mb  = sqrtf(gxb * gxb + gyb * gyb);

      float d = ma - mb;
      acc += sqrtf(d * d + 1e-6f);
    }
  }

  // All 256 threads converged here -> EXEC all-1s for WMMA.
  float total = block_reduce_256(acc, red);
  if (tid == 0) {
    partial[(blockIdx.z * gridDim.y + blockIdx.y) * gridDim.x + blockIdx.x] = total;
  }
}

__global__ void __launch_bounds__(256)
sobel_finish_kernel(const float* __restrict__ partial,
                    float* __restrict__ out, int n) {
  __shared__ float red[128];
  float acc = 0.0f;
  for (int i = threadIdx.x; i < n; i += 256) acc += partial[i];
  float total = block_reduce_256(acc, red);
  if (threadIdx.x == 0) out[0] = total * (1.0f / 16.0f);  // / batch (B=16)
}

extern "C" void kernel_launch(void* const* d_in, const int* in_sizes, int n_in,
                              void* d_out, int out_size, void* d_ws, size_t ws_size,
                              hipStream_t stream) {
  (void)in_sizes; (void)n_in; (void)out_size; (void)ws_size;
  const float* src = (const float*)d_in[0];
  const float* tgt = (const float*)d_in[1];
  float* out     = (float*)d_out;
  float* partial = (float*)d_ws;   // NBLOCKS floats = 12 KB of scratch

  dim3 grid(4, 16, 48);            // ceil(510/128), ceil(510/32), B*C
  sobel_partial_kernel<<<grid, 256, 0, stream>>>(src, tgt, partial);
  sobel_finish_kernel<<<1, 256, 0, stream>>>(partial, out, NBLOCKS);
}